// Q_83305185673803
// MI455X (gfx1250) — compile-verified
//
#include <hip/hip_runtime.h>

#define N_NODES 16384
#define DIM 64

typedef float f4  __attribute__((ext_vector_type(4)));
typedef float v2f __attribute__((ext_vector_type(2)));
typedef float v8f __attribute__((ext_vector_type(8)));

// ---------------------------------------------------------------------------
// K1: streaming row reduction over graph (1 GiB, the whole roofline).
// pos[i] = sum_j relu(graph[i,j]), neg[i] = sum_j relu(-graph[i,j])
// ---------------------------------------------------------------------------
__global__ void rowsum_kernel(const float* __restrict__ graph,
                              float* __restrict__ pos,
                              float* __restrict__ neg) {
  __shared__ float sp[256];
  __shared__ float sn[256];
  const int row = blockIdx.x;
  const int tid = threadIdx.x;
  const f4* rp = (const f4*)(graph + (size_t)row * N_NODES);
  float p = 0.f, n = 0.f;
  for (int i = tid; i < N_NODES / 4; i += 256) {
    f4 x = __builtin_nontemporal_load(rp + i);   // NT: no reuse, 1 GiB stream
    p += fmaxf(x.x, 0.f) + fmaxf(x.y, 0.f) + fmaxf(x.z, 0.f) + fmaxf(x.w, 0.f);
    n += fmaxf(-x.x, 0.f) + fmaxf(-x.y, 0.f) + fmaxf(-x.z, 0.f) + fmaxf(-x.w, 0.f);
  }
  sp[tid] = p; sn[tid] = n;
  __syncthreads();
  for (int s = 128; s > 0; s >>= 1) {
    if (tid < s) { sp[tid] += sp[tid + s]; sn[tid] += sn[tid + s]; }
    __syncthreads();
  }
  if (tid == 0) { pos[row] = sp[0]; neg[row] = sn[0]; }
}

// ---------------------------------------------------------------------------
// K2: a = W3 @ relu(w4), b = W3 @ relu(-w4)   (64-vectors; rank-2 collapse
// of edge_msg, so no [N,64]x[64,64] GEMM is needed for `base`)
// ---------------------------------------------------------------------------
__global__ void vecab_kernel(const float* __restrict__ W3,
                             const float* __restrict__ w4,
                             float* __restrict__ a, float* __restrict__ b) {
  const int d = threadIdx.x;  // 64 threads
  float av = 0.f, bv = 0.f;
  for (int k = 0; k < DIM; ++k) {
    const float w = W3[d * DIM + k];
    av += w * fmaxf(w4[k], 0.f);
    bv += w * fmaxf(-w4[k], 0.f);
  }
  a[d] = av; b[d] = bv;
}

// ---------------------------------------------------------------------------
// K3: base[i,d] = x[i]*w1[d] + pos[i]*a[d] + neg[i]*b[d]
//     u1 = relu(base)   (iteration 1 has summ=0, u0=0 -> GEMM skipped)
// ---------------------------------------------------------------------------
__global__ void base_kernel(const int* __restrict__ s_mask,
                            const float* __restrict__ w1,
                            const float* __restrict__ pos,
                            const float* __restrict__ neg,
                            const float* __restrict__ a,
                            const float* __restrict__ b,
                            float* __restrict__ base,
                            float* __restrict__ u) {
  const int idx = blockIdx.x * 256 + threadIdx.x;
  const int i = idx >> 6;
  const int d = idx & 63;
  const float x = (float)s_mask[i];
  const float bs = x * w1[d] + pos[i] * a[d] + neg[i] * b[d];
  base[idx] = bs;
  u[idx] = fmaxf(bs, 0.f);
}

// ---------------------------------------------------------------------------
// K4a: summ[d] = sum_i u[i,d]   (4 MiB read, entirely L2-resident)
// ---------------------------------------------------------------------------
__global__ void colsum_kernel(const float* __restrict__ u,
                              float* __restrict__ summ) {
  __shared__ float s[256];
  const int d = blockIdx.x;     // 64 blocks, one per feature
  const int tid = threadIdx.x;  // 256 threads
  float acc = 0.f;
  for (int i = tid; i < N_NODES; i += 256) acc += u[(size_t)i * DIM + d];
  s[tid] = acc;
  __syncthreads();
  for (int k = 128; k > 0; k >>= 1) {
    if (tid < k) s[tid] += s[tid + k];
    __syncthreads();
  }
  if (tid == 0) summ[d] = s[0];
}

// ---------------------------------------------------------------------------
// K4b: u_new[i,:] = relu(base[i,:] + (summ - u[i,:]) @ W2^T)  via f32 WMMA.
// Each wave: one 16-row tile, 4 accumulators (16x16 f32, C preloaded = base),
// 16 chained V_WMMA_F32_16X16X4_F32 k-steps per accumulator.
// A 16x4 layout (ISA 7.12.2): lane = M (mod 16); V0 K = k0+2*hi, V1 K = +1.
// C/D layout: VGPR r -> M = r + 8*hi, lane%16 -> N.
// ---------------------------------------------------------------------------
__global__ void wmma_step_kernel(const float* __restrict__ base,
                                 float* __restrict__ u,
                                 const float* __restrict__ summ,
                                 const float* __restrict__ W2) {
  const int wave = threadIdx.x >> 5;       // 4 waves / block
  const int lane = threadIdx.x & 31;
  const int l16  = lane & 15;
  const int hi   = lane >> 4;
  const int row0 = blockIdx.x * 64 + wave * 16;

  v8f acc[4];
  // Preload C accumulators with base tile (bias-add for free).
  for (int nt = 0; nt < 4; ++nt)
    for (int r = 0; r < 8; ++r)
      acc[nt][r] = base[(size_t)(row0 + r + 8 * hi) * DIM + nt * 16 + l16];

  for (int kk = 0; kk < 16; ++kk) {
    const int ka = kk * 4 + 2 * hi;
    v2f afrag;
    afrag.x = summ[ka]     - u[(size_t)(row0 + l16) * DIM + ka];
    afrag.y = summ[ka + 1] - u[(size_t)(row0 + l16) * DIM + ka + 1];
    for (int nt = 0; nt < 4; ++nt) {
      v2f bfrag;  // B[k,n] = W2[n,k]  (computing s @ W2^T)
      bfrag.x = W2[(nt * 16 + l16) * DIM + ka];
      bfrag.y = W2[(nt * 16 + l16) * DIM + ka + 1];
      acc[nt] = __builtin_amdgcn_wmma_f32_16x16x4_f32(
          false, afrag, false, bfrag, (short)0, acc[nt], false, false);
    }
  }

  // relu + in-place store (each wave only touches its own 16 rows,
  // and all its u reads precede its writes in program order)
  for (int nt = 0; nt < 4; ++nt)
    for (int r = 0; r < 8; ++r)
      u[(size_t)(row0 + r + 8 * hi) * DIM + nt * 16 + l16] =
          fmaxf(acc[nt][r], 0.f);
}

// ---------------------------------------------------------------------------
// K5: out = w5 . relu(cat(W6 @ summ, W7 @ u[v]))
// ---------------------------------------------------------------------------
__global__ void final_kernel(const float* __restrict__ u,
                             const float* __restrict__ summ,
                             const int* __restrict__ vptr,
                             const float* __restrict__ W6,
                             const float* __restrict__ W7,
                             const float* __restrict__ w5,
                             float* __restrict__ out) {
  __shared__ float s[128];
  const int t = threadIdx.x;  // 128 threads
  const int v = vptr[0];
  float acc = 0.f;
  if (t < 64) {
    for (int k = 0; k < DIM; ++k) acc += summ[k] * W6[t * DIM + k];
  } else {
    const int d = t - 64;
    for (int k = 0; k < DIM; ++k) acc += u[(size_t)v * DIM + k] * W7[d * DIM + k];
  }
  s[t] = fmaxf(acc, 0.f) * w5[t];
  __syncthreads();
  for (int k = 64; k > 0; k >>= 1) {
    if (t < k) s[t] += s[t + k];
    __syncthreads();
  }
  if (t == 0) out[0] = s[0];
}

// ---------------------------------------------------------------------------
extern "C" void kernel_launch(void* const* d_in, const int* in_sizes, int n_in,
                              void* d_out, int out_size, void* d_ws, size_t ws_size,
                              hipStream_t stream) {
  const float* graph  = (const float*)d_in[0];
  const int*   s_mask = (const int*)  d_in[1];
  const int*   vptr   = (const int*)  d_in[2];
  const float* w1     = (const float*)d_in[3];
  const float* W2     = (const float*)d_in[4];
  const float* W3     = (const float*)d_in[5];
  const float* w4     = (const float*)d_in[6];
  const float* w5     = (const float*)d_in[7];
  const float* W6     = (const float*)d_in[8];
  const float* W7     = (const float*)d_in[9];
  float* out = (float*)d_out;

  // Workspace carve-up (floats): pos[N] neg[N] a[64] b[64] summ[64] pad | base[N*64] u[N*64]
  float* ws   = (float*)d_ws;
  float* pos  = ws;
  float* neg  = ws + N_NODES;
  float* a    = ws + 2 * N_NODES;
  float* b    = a + 64;
  float* summ = b + 64;
  float* base = ws + 2 * N_NODES + 256;
  float* u    = base + (size_t)N_NODES * DIM;

  // 1) pos/neg row sums over graph (the 1 GiB stream — dominates runtime)
  rowsum_kernel<<<N_NODES, 256, 0, stream>>>(graph, pos, neg);
  // 2) collapse edge_msg to rank-2 vectors
  vecab_kernel<<<1, 64, 0, stream>>>(W3, w4, a, b);
  // 3) base + iteration 1 (u = relu(base))
  base_kernel<<<(N_NODES * DIM) / 256, 256, 0, stream>>>(
      s_mask, w1, pos, neg, a, b, base, u);
  // 4) iterations 2..4: reduce then WMMA step
  for (int it = 0; it < 3; ++it) {
    colsum_kernel<<<DIM, 256, 0, stream>>>(u, summ);
    wmma_step_kernel<<<N_NODES / 64, 128, 0, stream>>>(base, u, summ, W2);
  }
  // 5) final reduction of u, then readout
  colsum_kernel<<<DIM, 256, 0, stream>>>(u, summ);
  final_kernel<<<1, 128, 0, stream>>>(u, summ, vptr, W6, W7, w5, out);
}